// MultiHeadAttention_88862873354527
// MI455X (gfx1250) — compile-verified
//
#include <hip/hip_runtime.h>
#include <cstdint>

typedef float v2f __attribute__((ext_vector_type(2)));
typedef float v8f __attribute__((ext_vector_type(8)));

namespace {
constexpr int HEADS = 8;
constexpr int B_    = 8;
constexpr int I_    = 2048;
constexpr int J_    = 512;
constexpr int DIM_  = 256;
constexpr int DH    = 32;     // head dim
constexpr int ITILE = 64;     // query rows per workgroup
constexpr int NW    = 4;      // wave32s per workgroup
constexpr int KVS   = 34;     // LDS row stride (floats) for K' rows (even -> b64 aligned, conflict-free)
constexpr int VTS   = 516;    // LDS row stride (floats) for transposed V (d-major)
constexpr int SST   = 516;    // LDS row stride (floats) for S/P strip
constexpr float SCALE = 0.17677669529663687f;  // 32^-0.5
}

__global__ __launch_bounds__(128, 1)
void mha_rpe_fused(const float* __restrict__ Q, const float* __restrict__ K,
                   const float* __restrict__ V, const float* __restrict__ rpe,
                   float* __restrict__ out, float* __restrict__ dots)
{
  extern __shared__ float smem[];
  float* sK   = smem;                    // J_*KVS    : K + rpe (per head), row-major [j][d]
  float* sVt  = sK + J_ * KVS;           // DH*VTS    : V transposed, [d][j]
  float* sS   = sVt + DH * VTS;          // NW*16*SST : per-wave S/P strip
  float* sInv = sS + NW * 16 * SST;      // NW*16     : 1/rowsum

  const int tid = threadIdx.x;
  const int wg  = blockIdx.x;
  const int ib  = wg & (I_ / ITILE - 1);          // 32 i-blocks
  const int h   = (wg >> 5) & (HEADS - 1);
  const int b   = wg >> 8;
  const int i0  = ib * ITILE;

  // ---------- Phase 0: stage K' = K + rpe (row-major) and V (transposed) ----------
  for (int idx = tid; idx < J_ * (DH / 4); idx += 128) {
    const int j  = idx >> 3;
    const int kd = (idx & 7) * 4;
    const float4 kv = *(const float4*)(K   + ((size_t)b * J_ + j) * DIM_ + h * DH + kd);
    const float4 rv = *(const float4*)(rpe + ((size_t)h * J_ + j) * DH + kd);
    const float4 vv = *(const float4*)(V   + ((size_t)b * J_ + j) * DIM_ + h * DH + kd);
    float* dk = sK + j * KVS + kd;
    dk[0] = kv.x + rv.x; dk[1] = kv.y + rv.y; dk[2] = kv.z + rv.z; dk[3] = kv.w + rv.w;
    float* dv = sVt + kd * VTS + j;               // transpose on the way in
    dv[0 * VTS] = vv.x; dv[1 * VTS] = vv.y; dv[2 * VTS] = vv.z; dv[3 * VTS] = vv.w;
  }
  __syncthreads();

  const int wave = tid >> 5;
  const int lane = tid & 31;
  const int m    = lane & 15;   // A-matrix row / B-matrix col / C-matrix col
  const int hh   = lane >> 4;   // lane-half selects upper K / upper M

  // ---------- Load Q fragments (scale folded in) ----------
  // A 16x4 f32 layout: M = lane%16, K = vgpr + 2*(lane/16)
  v2f qf[8];
  const float* qb = Q + ((size_t)b * I_ + (i0 + wave * 16 + m)) * DIM_ + h * DH;
  #pragma unroll
  for (int s = 0; s < 8; ++s) {
    v2f t = *(const v2f*)(qb + 4 * s + 2 * hh);
    qf[s].x = t.x * SCALE;
    qf[s].y = t.y * SCALE;
  }

  float* sSw = sS + wave * 16 * SST;

  // Per-lane base pointers; the 8 C-rows live at compile-time-constant offsets.
  float* dotsP = dots + (((size_t)b * HEADS + h) * I_ + (i0 + wave * 16 + 8 * hh)) * (size_t)J_ + m;
  float* sSp   = sSw + (8 * hh) * SST + m;
  const float* kB = sK + (size_t)m * KVS + 2 * hh;

  // ---------- Phase 1: S = Qs @ K'^T  (write to dots [NT] + LDS strip) ----------
  for (int jt = 0; jt < J_ / 16; ++jt) {
    const int j0 = jt * 16;
    v8f c = {};
    #pragma unroll
    for (int s = 0; s < 8; ++s) {
      // B 4x16 layout: N = lane%16 (j), K = vgpr + 2*(lane/16) (d); B[k][n] = K'[j0+n][4s+k]
      v2f bf = *(const v2f*)(kB + (size_t)j0 * KVS + 4 * s);
      c = __builtin_amdgcn_wmma_f32_16x16x4_f32(false, qf[s], false, bf,
                                                (short)0, c, false, false);
    }
    #pragma unroll
    for (int v = 0; v < 8; ++v) {             // C/D: M = vgpr + 8*(lane/16), N = lane%16
      __builtin_nontemporal_store(c[v], dotsP + (size_t)v * J_ + j0);
      sSp[v * SST + j0] = c[v];
    }
  }

  // ---------- Phase 2: row softmax (keep unnormalized exp; save 1/sum) ----------
  {
    const int row = lane >> 1;
    float* sr = sSw + row * SST + (lane & 1) * (J_ / 2);
    float mx = -INFINITY;
    for (int j = 0; j < J_ / 2; j += 4) {
      float4 t = *(const float4*)(sr + j);
      mx = fmaxf(mx, fmaxf(fmaxf(t.x, t.y), fmaxf(t.z, t.w)));
    }
    mx = fmaxf(mx, __shfl_xor(mx, 1, 32));
    float sum = 0.0f;
    for (int j = 0; j < J_ / 2; j += 4) {
      float4 t = *(const float4*)(sr + j);
      t.x = __expf(t.x - mx); t.y = __expf(t.y - mx);
      t.z = __expf(t.z - mx); t.w = __expf(t.w - mx);
      sum += (t.x + t.y) + (t.z + t.w);
      *(float4*)(sr + j) = t;
    }
    sum += __shfl_xor(sum, 1, 32);
    if ((lane & 1) == 0) sInv[wave * 16 + row] = 1.0f / sum;
  }

  // ---------- Phase 3: O = P @ V  (normalize in epilogue, NT store) ----------
  const float* aB = sSw + (size_t)m * SST + 2 * hh;            // A: P[m][j0 + v + 2hh]
  float* outP = out + ((size_t)b * I_ + (i0 + wave * 16 + 8 * hh)) * DIM_ + h * DH + m;
  #pragma unroll
  for (int nt = 0; nt < 2; ++nt) {
    const int n0 = nt * 16;
    const float* vB = sVt + (size_t)(n0 + m) * VTS + 2 * hh;   // B: Vt[n0+m][j0 + v + 2hh]
    v8f c = {};
    #pragma unroll 4
    for (int k = 0; k < J_ / 4; ++k) {
      const int j0 = k * 4;
      v2f af = *(const v2f*)(aB + j0);
      v2f bf = *(const v2f*)(vB + j0);
      c = __builtin_amdgcn_wmma_f32_16x16x4_f32(false, af, false, bf,
                                                (short)0, c, false, false);
    }
    #pragma unroll
    for (int v = 0; v < 8; ++v) {
      const int Mi = v + 8 * hh;
      __builtin_nontemporal_store(c[v] * sInv[wave * 16 + Mi], outP + (size_t)v * DIM_ + n0);
    }
  }
}

extern "C" void kernel_launch(void* const* d_in, const int* in_sizes, int n_in,
                              void* d_out, int out_size, void* d_ws, size_t ws_size,
                              hipStream_t stream) {
  (void)in_sizes; (void)n_in; (void)out_size; (void)d_ws; (void)ws_size;
  const float* Q   = (const float*)d_in[0];
  const float* K   = (const float*)d_in[1];
  const float* V   = (const float*)d_in[2];
  const float* rpe = (const float*)d_in[3];
  float* out  = (float*)d_out;
  float* dots = out + (size_t)B_ * I_ * DIM_;   // tuple order: (out, dots)

  const size_t shmem = (size_t)(J_ * KVS + DH * VTS + NW * 16 * SST + NW * 16) * sizeof(float);
  dim3 grid(B_ * HEADS * (I_ / ITILE));   // 2048 workgroups
  dim3 block(128);                        // 4 wave32s
  mha_rpe_fused<<<grid, block, shmem, stream>>>(Q, K, V, rpe, out, dots);
}